// serverNewsModel_18433999635116
// MI455X (gfx1250) — compile-verified
//
#include <hip/hip_runtime.h>
#include <hip/hip_bf16.h>
#include <math.h>

typedef float v2f __attribute__((ext_vector_type(2)));
typedef float v8f __attribute__((ext_vector_type(8)));

#define NFEAT 256
#define NHID  64
#define NEG_SLOPE 0.2f

// ---------------------------------------------------------------------------
// K0: init out = bias (atomics accumulate on top), m = -inf, denom = 0
// ---------------------------------------------------------------------------
__global__ void gat_init(float* __restrict__ out, const float* __restrict__ bias,
                         float* __restrict__ mmax, float* __restrict__ denom, int n) {
    int i = blockIdx.x * blockDim.x + threadIdx.x;
    if (i < n * NHID) out[i] = bias[i & (NHID - 1)];
    if (i < n) { mmax[i] = -3.0e38f; denom[i] = 0.0f; }
}

// ---------------------------------------------------------------------------
// K1: xp = x @ W via V_WMMA_F32_16X16X4_F32 (full f32 precision).
// Each wave computes a 16x64 strip of xp (4 accumulators of v8f).
// A fragment (16x4 f32): lanes 0-15 row M=lane {K=k,k+1}, lanes 16-31 {K=k+2,k+3}
// B fragment (4x16 f32): lanes 0-15 col N=lane {K=k,k+1}, lanes 16-31 {K=k+2,k+3}
// ---------------------------------------------------------------------------
__global__ void __launch_bounds__(256)
gat_gemm(const float* __restrict__ X, const float* __restrict__ W,
         float* __restrict__ XP, int n) {
    const int tid  = threadIdx.x;
    const int wave = tid >> 5;
    const int lane = tid & 31;
    const int half = lane >> 4;   // 0: lanes 0-15, 1: lanes 16-31
    const int lrow = lane & 15;
    const int r0   = (blockIdx.x * 8 + wave) * 16;
    if (r0 >= n) return;          // wave-uniform exit: EXEC stays all-1s for WMMA

    int arow = r0 + lrow;
    if (arow > n - 1) arow = n - 1;     // clamp loads; stores are row-guarded
    const float* aptr = X + (size_t)arow * NFEAT + 2 * half;

    v8f c0 = {}, c1 = {}, c2 = {}, c3 = {};

#pragma unroll 8
    for (int k = 0; k < NFEAT; k += 4) {
        v2f a = *(const v2f*)(aptr + k);                    // 8B load, 8B aligned
        const float* wp = W + (size_t)(k + 2 * half) * NHID + lrow;
        v2f b0, b1, b2, b3;
        b0[0] = wp[0];   b0[1] = wp[NHID + 0];
        b1[0] = wp[16];  b1[1] = wp[NHID + 16];
        b2[0] = wp[32];  b2[1] = wp[NHID + 32];
        b3[0] = wp[48];  b3[1] = wp[NHID + 48];
        c0 = __builtin_amdgcn_wmma_f32_16x16x4_f32(false, a, false, b0, (short)0, c0, false, false);
        c1 = __builtin_amdgcn_wmma_f32_16x16x4_f32(false, a, false, b1, (short)0, c1, false, false);
        c2 = __builtin_amdgcn_wmma_f32_16x16x4_f32(false, a, false, b2, (short)0, c2, false, false);
        c3 = __builtin_amdgcn_wmma_f32_16x16x4_f32(false, a, false, b3, (short)0, c3, false, false);
    }

    // C/D layout: lane group selects N=lrow; VGPR j holds row M = j + 8*half
#pragma unroll
    for (int j = 0; j < 8; ++j) {
        int row = half * 8 + j;
        if (r0 + row < n) {
            float* o = XP + (size_t)(r0 + row) * NHID + lrow;
            o[0]  = c0[j];
            o[16] = c1[j];
            o[32] = c2[j];
            o[48] = c3[j];
        }
    }
}

// ---------------------------------------------------------------------------
// K2: per-node scores s_src = xp . a_src, s_dst = xp . a_dst
// 16 lanes per node, float4 loads (256B coalesced per node), shfl_xor reduce
// ---------------------------------------------------------------------------
__global__ void __launch_bounds__(256)
gat_scores(const float* __restrict__ XP, const float* __restrict__ As,
           const float* __restrict__ Ad, float* __restrict__ Ss,
           float* __restrict__ Sd, int n) {
    const int tid  = threadIdx.x;
    const int node = blockIdx.x * 16 + (tid >> 4);
    const int part = tid & 15;
    if (node >= n) return;
    float4 xv = ((const float4*)(XP + (size_t)node * NHID))[part];
    float4 av = ((const float4*)As)[part];
    float4 dv = ((const float4*)Ad)[part];
    float ss = xv.x * av.x + xv.y * av.y + xv.z * av.z + xv.w * av.w;
    float sd = xv.x * dv.x + xv.y * dv.y + xv.z * dv.z + xv.w * dv.w;
#pragma unroll
    for (int m = 1; m < 16; m <<= 1) {   // masks 1..8 stay inside the 16-lane half
        ss += __shfl_xor(ss, m, 32);
        sd += __shfl_xor(sd, m, 32);
    }
    if (part == 0) { Ss[node] = ss; Sd[node] = sd; }
}

// ---------------------------------------------------------------------------
// K3: per-edge logits e = leaky_relu(s_src[j] + s_dst[i]); segment max via
// atomicMax. Edge ids [E, E+N) are the implicit self-loops.
// ---------------------------------------------------------------------------
__global__ void gat_logits(const int* __restrict__ src, const int* __restrict__ dst,
                           const float* __restrict__ Ss, const float* __restrict__ Sd,
                           float* __restrict__ ew, float* __restrict__ mmax,
                           int E, int n) {
    int k = blockIdx.x * blockDim.x + threadIdx.x;
    if (k >= E + n) return;
    int s = (k < E) ? src[k] : (k - E);
    int d = (k < E) ? dst[k] : (k - E);
    float v = Ss[s] + Sd[d];
    float e = (v > 0.0f) ? v : NEG_SLOPE * v;
    ew[k] = e;
    atomicMax(mmax + d, e);
}

// ---------------------------------------------------------------------------
// K4: ex = exp(e - m[dst]); denom[dst] += ex
// ---------------------------------------------------------------------------
__global__ void gat_expsum(const int* __restrict__ dst, float* __restrict__ ew,
                           const float* __restrict__ mmax, float* __restrict__ denom,
                           int E, int n) {
    int k = blockIdx.x * blockDim.x + threadIdx.x;
    if (k >= E + n) return;
    int d = (k < E) ? dst[k] : (k - E);
    float ex = __expf(ew[k] - mmax[d]);
    ew[k] = ex;
    atomicAdd(denom + d, ex);
}

// ---------------------------------------------------------------------------
// K5: out[dst] += (ex/denom[dst]) * xp[src].  16 lanes per edge, float4 gather
// (256B coalesced per edge), 4 f32 atomics per lane into L2-resident output.
// ---------------------------------------------------------------------------
__global__ void __launch_bounds__(256)
gat_scatter(const int* __restrict__ src, const int* __restrict__ dst,
            const float* __restrict__ ew, const float* __restrict__ denom,
            const float* __restrict__ XP, float* __restrict__ out, int E, int n) {
    const int tid  = threadIdx.x;
    const int k    = blockIdx.x * 16 + (tid >> 4);
    if (k >= E + n) return;
    const int part = tid & 15;
    int s = (k < E) ? src[k] : (k - E);
    int d = (k < E) ? dst[k] : (k - E);
    float alpha = ew[k] / denom[d];
    float4 xv = ((const float4*)(XP + (size_t)s * NHID))[part];
    float* o = out + (size_t)d * NHID + part * 4;
    atomicAdd(o + 0, alpha * xv.x);
    atomicAdd(o + 1, alpha * xv.y);
    atomicAdd(o + 2, alpha * xv.z);
    atomicAdd(o + 3, alpha * xv.w);
}

// ---------------------------------------------------------------------------
extern "C" void kernel_launch(void* const* d_in, const int* in_sizes, int n_in,
                              void* d_out, int out_size, void* d_ws, size_t ws_size,
                              hipStream_t stream) {
    const float* x     = (const float*)d_in[0];
    const int*   ei    = (const int*)  d_in[1];
    const float* W     = (const float*)d_in[2];
    const float* a_src = (const float*)d_in[3];
    const float* a_dst = (const float*)d_in[4];
    const float* bias  = (const float*)d_in[5];
    float*       out   = (float*)d_out;

    const int N = in_sizes[0] / NFEAT;   // 100000
    const int E = in_sizes[1] / 2;       // 1600000
    const int* src = ei;
    const int* dst = ei + E;

    // workspace carve-up (all f32)
    float* xp    = (float*)d_ws;               // N*64
    float* ew    = xp    + (size_t)N * NHID;   // E+N
    float* mmax  = ew    + (size_t)(E + N);    // N
    float* denom = mmax  + N;                  // N
    float* ssrc  = denom + N;                  // N
    float* sdst  = ssrc  + N;                  // N

    gat_init   <<<(N * NHID + 255) / 256, 256, 0, stream>>>(out, bias, mmax, denom, N);
    gat_gemm   <<<(N + 127) / 128,        256, 0, stream>>>(x, W, xp, N);
    gat_scores <<<(N + 15) / 16,          256, 0, stream>>>(xp, a_src, a_dst, ssrc, sdst, N);
    gat_logits <<<(E + N + 255) / 256,    256, 0, stream>>>(src, dst, ssrc, sdst, ew, mmax, E, N);
    gat_expsum <<<(E + N + 255) / 256,    256, 0, stream>>>(dst, ew, mmax, denom, E, N);
    gat_scatter<<<(E + N + 15) / 16,      256, 0, stream>>>(src, dst, ew, denom, xp, out, E, N);
}